// DedicomDecoder_43215960932829
// MI455X (gfx1250) — compile-verified
//
#include <hip/hip_runtime.h>
#include <hip/hip_bf16.h>

#define N_DRUGS 4096
#define W_DIM   512
#define N_EDGES 131072

typedef __attribute__((ext_vector_type(2))) float v2f;
typedef __attribute__((ext_vector_type(8))) float v8f;

// ---------------------------------------------------------------------------
// Stage 1: Y[m][n] = d[n] * sum_k (Z[m][k] * d[k]) * W[k][n]
// FP32 WMMA (V_WMMA_F32_16X16X4_F32), block tile 128x64, BK=32 via LDS.
// B tile stored k-pair-interleaved so each B fragment is ONE ds_load_b64
// into an even-aligned VGPR pair (no repacking movs before v_wmma).
// ---------------------------------------------------------------------------
__global__ __launch_bounds__(256)
void dedicom_gemm(const float* __restrict__ Z,
                  const float* __restrict__ W,
                  const float* __restrict__ local_diag,
                  const int*   __restrict__ sub_idx,
                  float* __restrict__ Y) {
    const int tid   = threadIdx.x;
    const int lane  = tid & 31;
    const int wave  = tid >> 5;
    const int row0  = blockIdx.y * 128;      // M tile base
    const int n0g   = blockIdx.x * 64;       // N tile base
    const int m0    = wave * 16;             // this wave's 16 rows inside tile
    const int mlane = lane & 15;
    const int nlane = lane & 15;
    const int koff  = (lane < 16) ? 0 : 2;   // ISA f32 A/B fragment half-wave K offset

    const float* __restrict__ dg = local_diag + (size_t)sub_idx[0] * W_DIM;

    // A tile: stride 36 (16B-aligned float4 staging; m*36%64 all-distinct banks)
    __shared__ float Zs[128 * 36];
    // B tile: 16 k-pairs, pair-interleaved; stride 160 (mod 64 == 32 -> the hi
    // half-wave's next-pair row hits a disjoint bank set; conflict-free b64)
    __shared__ float Ws[16 * 160];

    v8f acc[4];
    #pragma unroll
    for (int s = 0; s < 4; ++s)
        acc[s] = v8f{0.f, 0.f, 0.f, 0.f, 0.f, 0.f, 0.f, 0.f};

    for (int k0 = 0; k0 < W_DIM; k0 += 32) {
        __syncthreads();
        // ---- stage A tile: Zs[r][k] = Z[row0+r][k0+k] * d[k0+k]  (128x32) ----
        #pragma unroll
        for (int i = 0; i < 4; ++i) {
            const int idx = tid + 256 * i;        // float4 slot, 0..1023
            const int r   = idx >> 3;             // 0..127
            const int c4  = (idx & 7) << 2;       // 0..28
            const float4 z4 = *(const float4*)&Z[(size_t)(row0 + r) * W_DIM + k0 + c4];
            const float4 d4 = *(const float4*)&dg[k0 + c4];
            *(float4*)&Zs[r * 36 + c4] =
                make_float4(z4.x * d4.x, z4.y * d4.y, z4.z * d4.z, z4.w * d4.w);
        }
        // ---- stage B tile, k-pair interleaved: Ws[kp*160 + 2n + j] = W[k0+2kp+j][n0g+n] ----
        {
            const int kp = tid >> 4;              // 0..15
            const int n4 = (tid & 15) << 2;       // 0..60
            const float4 w0 = *(const float4*)&W[(size_t)(k0 + 2 * kp    ) * W_DIM + n0g + n4];
            const float4 w1 = *(const float4*)&W[(size_t)(k0 + 2 * kp + 1) * W_DIM + n0g + n4];
            float* dst = &Ws[kp * 160 + 2 * n4];  // 32B-aligned
            *(float4*)(dst + 0) = make_float4(w0.x, w1.x, w0.y, w1.y);
            *(float4*)(dst + 4) = make_float4(w0.z, w1.z, w0.w, w1.w);
        }
        // prefetch next K chunk (global_prefetch_b8)
        if (k0 + 32 < W_DIM) {
            __builtin_prefetch(&Z[(size_t)(row0 + (tid >> 3)) * W_DIM + k0 + 32], 0, 0);
            __builtin_prefetch(&W[(size_t)(k0 + 32 + 2 * (tid >> 4)) * W_DIM + n0g], 0, 0);
        }
        __syncthreads();

        // ---- compute: 8 k-steps of 4, 4 N-subtiles of 16 ----
        #pragma unroll
        for (int kk = 0; kk < 32; kk += 4) {
            // A frag: {A[m][kk+koff], A[m][kk+koff+1]}  (8B-aligned ds_load_b64)
            const v2f a = *(const v2f*)&Zs[(m0 + mlane) * 36 + kk + koff];
            const int kp = (kk + koff) >> 1;      // lanes<16: kk/2 ; lanes>=16: kk/2+1
            #pragma unroll
            for (int s = 0; s < 4; ++s) {
                // B frag: {W[kk+koff][n], W[kk+koff+1][n]} as one contiguous b64
                const v2f b = *(const v2f*)&Ws[kp * 160 + 2 * (s * 16 + nlane)];
                acc[s] = __builtin_amdgcn_wmma_f32_16x16x4_f32(
                    /*neg_a=*/false, a, /*neg_b=*/false, b,
                    /*c_mod=*/(short)0, acc[s],
                    /*reuse_a=*/false, /*reuse_b=*/false);
            }
        }
    }

    // ---- writeback with output-side diag scale; C/D layout: VGPR v -> M=v (+8 hi half) ----
    const int rbase = row0 + m0 + ((lane < 16) ? 0 : 8);
    #pragma unroll
    for (int s = 0; s < 4; ++s) {
        const int col = n0g + s * 16 + nlane;
        const float dn = dg[col];
        #pragma unroll
        for (int v = 0; v < 8; ++v) {
            Y[(size_t)(rbase + v) * W_DIM + col] = acc[s][v] * dn;
        }
    }
}

// ---------------------------------------------------------------------------
// Stage 2: per edge e, out[e] = sigmoid(dot(Y[row[e]], Z[col[e]]))
// One wave32 per edge, float4 loads, shuffle reduction. L2-resident gathers.
// ---------------------------------------------------------------------------
__global__ __launch_bounds__(256)
void dedicom_edge_score(const float* __restrict__ Y,
                        const float* __restrict__ Z,
                        const int*   __restrict__ edges,
                        float* __restrict__ out) {
    const int wave = threadIdx.x >> 5;
    const int lane = threadIdx.x & 31;
    const int e    = blockIdx.x * 8 + wave;
    if (e >= N_EDGES) return;

    const int r = edges[e];
    const int c = edges[N_EDGES + e];
    const float4* __restrict__ yr = (const float4*)(Y + (size_t)r * W_DIM);
    const float4* __restrict__ zc = (const float4*)(Z + (size_t)c * W_DIM);

    float s = 0.f;
    #pragma unroll
    for (int i = 0; i < 4; ++i) {          // 4 * 32 lanes * float4 = 512 floats
        const float4 a = yr[lane + 32 * i];
        const float4 b = zc[lane + 32 * i];
        s += a.x * b.x + a.y * b.y + a.z * b.z + a.w * b.w;
    }
    #pragma unroll
    for (int off = 16; off > 0; off >>= 1)
        s += __shfl_xor(s, off, 32);

    if (lane == 0)
        out[e] = 1.0f / (1.0f + __expf(-s));
}

// ---------------------------------------------------------------------------
extern "C" void kernel_launch(void* const* d_in, const int* in_sizes, int n_in,
                              void* d_out, int out_size, void* d_ws, size_t ws_size,
                              hipStream_t stream) {
    const float* z_drug     = (const float*)d_in[0];   // [4096, 512] f32
    const float* gweight    = (const float*)d_in[1];   // [512, 512] f32
    const float* local_diag = (const float*)d_in[2];   // [10, 512] f32
    const int*   edges      = (const int*)d_in[3];     // [2, 131072]
    const int*   sub_idx    = (const int*)d_in[4];     // scalar
    float*       out        = (float*)d_out;           // [131072] f32
    float*       Y          = (float*)d_ws;            // [4096, 512] f32 scratch (8 MB)

    dim3 gemm_grid(W_DIM / 64, N_DRUGS / 128);         // 8 x 32 blocks
    dedicom_gemm<<<gemm_grid, 256, 0, stream>>>(z_drug, gweight, local_diag, sub_idx, Y);

    dedicom_edge_score<<<N_EDGES / 8, 256, 0, stream>>>(Y, z_drug, edges, out);
}